// RandLANet_71725953843518
// MI455X (gfx1250) — compile-verified
//
#include <hip/hip_runtime.h>

// RandLANet gather/max/interp block for MI455X (gfx1250, wave32).
// Memory-movement op: gather sources (32MB feature / 32MB agg / 8MB pooled)
// all fit in the 192MB L2, so the gathers are L2-bandwidth-bound. Kernels are
// organized for max bytes/instruction (b128 gathers, 8 lanes x float4 per
// D=32 row, 4 rows per wave32), 16-deep load MLP, and a max3-friendly
// reduction tree (v_max3_num_f32) instead of a 15-deep serial max chain.

typedef __attribute__((ext_vector_type(4))) int   v4i;
typedef __attribute__((ext_vector_type(4))) float v4f;

static __device__ __forceinline__ v4f vmax2(v4f a, v4f b) {
    v4f r;
    r.x = fmaxf(a.x, b.x);
    r.y = fmaxf(a.y, b.y);
    r.z = fmaxf(a.z, b.z);
    r.w = fmaxf(a.w, b.w);
    return r;
}

// fmaxf(fmaxf(a,b),c) fuses to v_max3_num_f32 on gfx1250.
static __device__ __forceinline__ v4f vmax3(v4f a, v4f b, v4f c) {
    v4f r;
    r.x = fmaxf(fmaxf(a.x, b.x), c.x);
    r.y = fmaxf(fmaxf(a.y, b.y), c.y);
    r.z = fmaxf(fmaxf(a.z, b.z), c.z);
    r.w = fmaxf(fmaxf(a.w, b.w), c.w);
    return r;
}

// Problem constants (fixed by the reference).
constexpr int Bsz  = 4;
constexpr int Npts = 65536;   // N
constexpr int Nsub = 16384;   // NSUB
constexpr int K    = 16;
constexpr int D    = 32;      // floats per row; 8 lanes x float4
constexpr int BLK  = 256;     // threads per block (8 waves)

// Gather-max over K neighbors.
//   src: [Bsz, Npts, D] (batch stride Npts*D), idx: [rows, K], dst: [rows, D]
// Mapping: 8 lanes per row (lane&7 -> float4 slot), 4 rows per wave32.
template <int RPB_LOG2>
__global__ __launch_bounds__(BLK) void gather_max_kernel(
    const float* __restrict__ src,
    const int*   __restrict__ idx,
    float*       __restrict__ dst)
{
    const unsigned tid  = blockIdx.x * BLK + threadIdx.x;
    const unsigned lane = threadIdx.x & 31u;
    const unsigned sub  = lane >> 3;       // which of 4 rows in this wave
    const unsigned eoff = (lane & 7u) * 4; // float4 slot (element offset)
    const unsigned row  = (tid >> 5) * 4 + sub;
    const unsigned b    = row >> RPB_LOG2;

    const float* __restrict__ fb = src + (size_t)b * (Npts * D);
    const int*   __restrict__ ir = idx + (size_t)row * K;

    // Stream the 16 indices once (non-temporal: keep L2 for gathered rows).
    const v4i i0 = __builtin_nontemporal_load(reinterpret_cast<const v4i*>(ir + 0));
    const v4i i1 = __builtin_nontemporal_load(reinterpret_cast<const v4i*>(ir + 4));
    const v4i i2 = __builtin_nontemporal_load(reinterpret_cast<const v4i*>(ir + 8));
    const v4i i3 = __builtin_nontemporal_load(reinterpret_cast<const v4i*>(ir + 12));

    unsigned off[K];  // 32-bit element offsets into fb (max 2^21)
    off[0]  = (unsigned)i0.x * D + eoff;  off[1]  = (unsigned)i0.y * D + eoff;
    off[2]  = (unsigned)i0.z * D + eoff;  off[3]  = (unsigned)i0.w * D + eoff;
    off[4]  = (unsigned)i1.x * D + eoff;  off[5]  = (unsigned)i1.y * D + eoff;
    off[6]  = (unsigned)i1.z * D + eoff;  off[7]  = (unsigned)i1.w * D + eoff;
    off[8]  = (unsigned)i2.x * D + eoff;  off[9]  = (unsigned)i2.y * D + eoff;
    off[10] = (unsigned)i2.z * D + eoff;  off[11] = (unsigned)i2.w * D + eoff;
    off[12] = (unsigned)i3.x * D + eoff;  off[13] = (unsigned)i3.y * D + eoff;
    off[14] = (unsigned)i3.z * D + eoff;  off[15] = (unsigned)i3.w * D + eoff;

    // Warm the gather targets (gfx1250 global_prefetch_b8, no LOADcnt cost;
    // reuses the same computed offsets as the loads below).
#pragma unroll
    for (int k = 0; k < K; ++k) {
        __builtin_prefetch(fb + off[k], 0, 3);
    }

    // 16 independent b128 gathers in flight.
    v4f v[K];
#pragma unroll
    for (int k = 0; k < K; ++k) {
        v[k] = *reinterpret_cast<const v4f*>(fb + off[k]);
    }

    // 3-level max3 tree: 16 -> 5+1 -> 2 -> 1 (8 ops/component vs 15 serial).
    v4f a = vmax3(v[0],  v[1],  v[2]);
    v4f c = vmax3(v[3],  v[4],  v[5]);
    v4f d = vmax3(v[6],  v[7],  v[8]);
    v4f e = vmax3(v[9],  v[10], v[11]);
    v4f f = vmax3(v[12], v[13], v[14]);
    a = vmax3(a, c, v[15]);
    d = vmax3(d, e, f);
    const v4f m = vmax2(a, d);

    *reinterpret_cast<v4f*>(dst + (size_t)row * D + eoff) = m;
}

// out[row,:] = pooled[b, iidx[row], :] + out[row,:]   (out already holds agg)
__global__ __launch_bounds__(BLK) void interp_add_kernel(
    const float* __restrict__ pooled,
    const int*   __restrict__ iidx,
    float*       __restrict__ out)
{
    const unsigned tid  = blockIdx.x * BLK + threadIdx.x;
    const unsigned lane = threadIdx.x & 31u;
    const unsigned sub  = lane >> 3;
    const unsigned eoff = (lane & 7u) * 4;
    const unsigned row  = (tid >> 5) * 4 + sub;
    const unsigned b    = row >> 16;  // row / Npts

    const int ii = __builtin_nontemporal_load(iidx + row);

    const unsigned poff = (b * (unsigned)Nsub + (unsigned)ii) * D + eoff;
    const unsigned ooff = row * D + eoff;

    const v4f p = *reinterpret_cast<const v4f*>(pooled + poff);
    v4f a = *reinterpret_cast<const v4f*>(out + ooff);
    a.x += p.x; a.y += p.y; a.z += p.z; a.w += p.w;

    // Final result: stream-out, nobody re-reads it on-device.
    __builtin_nontemporal_store(a, reinterpret_cast<v4f*>(out + ooff));
}

extern "C" void kernel_launch(void* const* d_in, const int* in_sizes, int n_in,
                              void* d_out, int out_size, void* d_ws, size_t ws_size,
                              hipStream_t stream) {
    const float* feature = (const float*)d_in[0];  // [B, N, 1, D] == [B, N, D] flat
    const int*   nidx    = (const int*)  d_in[1];  // [B, N, K]
    const int*   pidx    = (const int*)  d_in[2];  // [B, NSUB, K]
    const int*   iidx    = (const int*)  d_in[3];  // [B, N, 1]

    float* out    = (float*)d_out;  // holds agg after k1, final result after k3
    float* pooled = (float*)d_ws;   // [B, NSUB, D] = 8 MiB scratch

    // 32 rows per 256-thread block (8 waves x 4 rows).
    constexpr int rows1 = Bsz * Npts;   // 262144
    constexpr int rows2 = Bsz * Nsub;   // 65536

    // 1) agg = max_k gather(feature, neighbor_idx)  -> d_out
    gather_max_kernel<16><<<rows1 / 32, BLK, 0, stream>>>(feature, nidx, out);
    // 2) pooled = max_k gather(agg, pool_idx)       -> d_ws
    gather_max_kernel<14><<<rows2 / 32, BLK, 0, stream>>>(out, pidx, pooled);
    // 3) out = gather(pooled, interp_idx) + agg     (in place on d_out)
    interp_add_kernel<<<rows1 / 32, BLK, 0, stream>>>(pooled, iidx, out);
}